// GATWithPooling_73787538145378
// MI455X (gfx1250) — compile-verified
//
#include <hip/hip_runtime.h>
#include <hip/hip_bf16.h>

typedef __attribute__((ext_vector_type(2))) float v2f;
typedef __attribute__((ext_vector_type(8))) float v8f;

#define NN 50000
#define EE 1600000
#define EP (EE + NN)      // edges + self loops
#define GG 64
#define DIN 128
#define HID 64
#define H1 4
#define F1 (H1 * HID)     // 256
#define F2 HID            // 64
#define NEG_SLOPE 0.2f
#define BN_EPS 1e-5f

#define CDIV(a, b) (((a) + (b) - 1) / (b))

// ---- monotonic float<->int mapping for atomicMax on floats ----
__device__ __forceinline__ int fmap(float f) {
    int i = __float_as_int(f);
    return i >= 0 ? i : (i ^ 0x7fffffff);
}
__device__ __forceinline__ float funmap(int i) {
    return __int_as_float(i >= 0 ? i : (i ^ 0x7fffffff));
}

// ---------------- generic fill ----------------
__global__ void fill_u32_kernel(unsigned* p, unsigned v, long long count) {
    long long i = (long long)blockIdx.x * blockDim.x + threadIdx.x;
    long long stride = (long long)gridDim.x * blockDim.x;
    for (; i < count; i += stride) p[i] = v;
}

// ---------------- fp32 GEMM via V_WMMA_F32_16X16X4_F32 ----------------
// C[M,Ncol] = A[M,K] * B[K,Ncol], all row-major fp32. One wave per 16x16 tile.
// A frag: lane L -> row = L%16, k = 2*(L/16)+{0,1}
// B frag: lane L -> col = L%16, k = 2*(L/16)+{0,1}
// C/D:    VGPR r, lane L -> row = r + 8*(L/16), col = L%16
__global__ void gemm16_wmma_f32(const float* __restrict__ A,
                                const float* __restrict__ B,
                                float* __restrict__ C,
                                int M, int K, int Ncol) {
    int wavesPerBlock = blockDim.x >> 5;
    int wid = blockIdx.x * wavesPerBlock + (threadIdx.x >> 5);
    int lane = threadIdx.x & 31;
    int ntN = Ncol >> 4;
    int numTiles = (M >> 4) * ntN;
    if (wid >= numTiles) return;                 // wave-uniform: EXEC all-ones inside
    int tm = wid / ntN, tn = wid % ntN;
    int r  = lane & 15;                          // A row / B,C col
    int kh = (lane >> 4) << 1;                   // 0 or 2

    const float* Arow = A + (size_t)(tm * 16 + r) * K;
    const float* Bcol = B + (tn * 16 + r);
    v8f acc = {};
    for (int k0 = 0; k0 < K; k0 += 4) {
        v2f a, b;
        a.x = Arow[k0 + kh];
        a.y = Arow[k0 + kh + 1];
        b.x = Bcol[(size_t)(k0 + kh) * Ncol];
        b.y = Bcol[(size_t)(k0 + kh + 1) * Ncol];
        acc = __builtin_amdgcn_wmma_f32_16x16x4_f32(
            /*neg_a=*/false, a, /*neg_b=*/false, b,
            /*c_mod=*/(short)0, acc, /*reuse_a=*/false, /*reuse_b=*/false);
    }
    float* Crow = C + (size_t)(tm * 16 + (lane >> 4) * 8) * Ncol + tn * 16 + r;
#pragma unroll
    for (int i = 0; i < 8; ++i) Crow[(size_t)i * Ncol] = acc[i];
}

// ---------------- per-node attention coefficients ----------------
template <int HEADS>
__global__ void alphas_kernel(const float* __restrict__ xw,
                              const float* __restrict__ att_s,
                              const float* __restrict__ att_d,
                              float* __restrict__ alpha_s,
                              float* __restrict__ alpha_d) {
    int t = blockIdx.x * blockDim.x + threadIdx.x;
    if (t >= NN * HEADS) return;
    int n = t / HEADS, h = t % HEADS;
    const float* row = xw + (size_t)n * HEADS * HID + h * HID;
    float ss = 0.f, sd = 0.f;
#pragma unroll 4
    for (int c = 0; c < HID; ++c) {
        float v = row[c];
        ss += v * att_s[h * HID + c];
        sd += v * att_d[h * HID + c];
    }
    alpha_s[t] = ss;
    alpha_d[t] = sd;
}

// ---------------- edge pass 1: segment max of leaky-relu scores ----------------
template <int HEADS>
__global__ void edge_max_kernel(const int* __restrict__ ei,
                                const float* __restrict__ as,
                                const float* __restrict__ ad,
                                int* __restrict__ m) {
    int t = blockIdx.x * blockDim.x + threadIdx.x;
    if (t >= EP * HEADS) return;
    int e = t / HEADS, h = t % HEADS;
    int s = e < EE ? ei[e] : e - EE;
    int d = e < EE ? ei[EE + e] : e - EE;
    float x = as[s * HEADS + h] + ad[d * HEADS + h];
    x = x >= 0.f ? x : NEG_SLOPE * x;
    atomicMax(&m[d * HEADS + h], fmap(x));
}

// ---------------- edge pass 2: exp & segment sum ----------------
template <int HEADS>
__global__ void edge_exp_kernel(const int* __restrict__ ei,
                                const float* __restrict__ as,
                                const float* __restrict__ ad,
                                const int* __restrict__ m,
                                float* __restrict__ den,
                                float* __restrict__ exw) {
    int t = blockIdx.x * blockDim.x + threadIdx.x;
    if (t >= EP * HEADS) return;
    int e = t / HEADS, h = t % HEADS;
    int s = e < EE ? ei[e] : e - EE;
    int d = e < EE ? ei[EE + e] : e - EE;
    float x = as[s * HEADS + h] + ad[d * HEADS + h];
    x = x >= 0.f ? x : NEG_SLOPE * x;
    float ex = __expf(x - funmap(m[d * HEADS + h]));
    exw[t] = ex;
    atomicAdd(&den[d * HEADS + h], ex);
}

// ---------------- edge pass 3: weighted scatter-add (one wave per edge) ----------------
template <int HEADS, int C>
__global__ void edge_agg_kernel(const int* __restrict__ ei,
                                const float* __restrict__ xw,
                                const float* __restrict__ exw,
                                const float* __restrict__ den,
                                float* __restrict__ agg) {
    int wid = (blockIdx.x * blockDim.x + threadIdx.x) >> 5;
    int lane = threadIdx.x & 31;
    if (wid >= EP) return;
    int s = wid < EE ? ei[wid] : wid - EE;
    int d = wid < EE ? ei[EE + wid] : wid - EE;
    const int F = HEADS * C;
    for (int c = lane; c < F; c += 32) {
        int h = c / C;
        float w = exw[(size_t)wid * HEADS + h] / (den[d * HEADS + h] + 1e-16f);
        atomicAdd(&agg[(size_t)d * F + c], xw[(size_t)s * F + c] * w);
    }
}

// ---------------- bias + eval batchnorm + relu (in place) ----------------
__global__ void bias_bn_relu_kernel(float* __restrict__ x,
                                    const float* __restrict__ bias,
                                    const float* __restrict__ g,
                                    const float* __restrict__ b,
                                    const float* __restrict__ mean,
                                    const float* __restrict__ var,
                                    int F, long long total) {
    long long i = (long long)blockIdx.x * blockDim.x + threadIdx.x;
    long long stride = (long long)gridDim.x * blockDim.x;
    for (; i < total; i += stride) {
        int f = (int)(i % F);
        float scale = g[f] * rsqrtf(var[f] + BN_EPS);
        float y = (x[i] + bias[f] - mean[f]) * scale + b[f];
        x[i] = fmaxf(y, 0.f);
    }
}

// ---------------- global mean pool (accumulate) ----------------
__global__ void pool_kernel(const float* __restrict__ h2,
                            const int* __restrict__ batch,
                            float* __restrict__ sums,
                            float* __restrict__ cnt) {
    int t = blockIdx.x * blockDim.x + threadIdx.x;
    if (t >= NN * F2) return;
    int n = t >> 6, f = t & 63;
    int b = batch[n];
    atomicAdd(&sums[b * F2 + f], h2[t]);
    if (f == 0) atomicAdd(&cnt[b], 1.0f);
}

// ---------------- classifier + log_softmax (tiny) ----------------
__global__ void classifier_kernel(const float* __restrict__ sums,
                                  const float* __restrict__ cnt,
                                  const float* __restrict__ Wc1,
                                  const float* __restrict__ bc1,
                                  const float* __restrict__ Wc2,
                                  const float* __restrict__ bc2,
                                  float* __restrict__ out) {
    int g = threadIdx.x;
    if (g >= GG) return;
    float inv = 1.0f / fmaxf(cnt[g], 1.0f);
    float gr[HID];
    for (int f = 0; f < HID; ++f) gr[f] = sums[g * HID + f] * inv;
    float l0 = bc2[0], l1 = bc2[1];
    for (int j = 0; j < HID; ++j) {
        float s = bc1[j];
        for (int f = 0; f < HID; ++f) s += gr[f] * Wc1[f * HID + j];
        float hc = fmaxf(s, 0.f);
        l0 += hc * Wc2[j * 2 + 0];
        l1 += hc * Wc2[j * 2 + 1];
    }
    float mx = fmaxf(l0, l1);
    float lse = mx + logf(expf(l0 - mx) + expf(l1 - mx));
    out[g * 2 + 0] = l0 - lse;
    out[g * 2 + 1] = l1 - lse;
}

extern "C" void kernel_launch(void* const* d_in, const int* in_sizes, int n_in,
                              void* d_out, int out_size, void* d_ws, size_t ws_size,
                              hipStream_t stream) {
    const float* x    = (const float*)d_in[0];
    const int*   ei   = (const int*)d_in[1];
    const int*   batch= (const int*)d_in[2];
    const float* W1   = (const float*)d_in[3];
    const float* as1w = (const float*)d_in[4];
    const float* ad1w = (const float*)d_in[5];
    const float* b1   = (const float*)d_in[6];
    const float* bn1g = (const float*)d_in[7];
    const float* bn1b = (const float*)d_in[8];
    const float* bn1m = (const float*)d_in[9];
    const float* bn1v = (const float*)d_in[10];
    const float* W2   = (const float*)d_in[11];
    const float* as2w = (const float*)d_in[12];
    const float* ad2w = (const float*)d_in[13];
    const float* b2   = (const float*)d_in[14];
    const float* bn2g = (const float*)d_in[15];
    const float* bn2b = (const float*)d_in[16];
    const float* bn2m = (const float*)d_in[17];
    const float* bn2v = (const float*)d_in[18];
    const float* Wc1  = (const float*)d_in[19];
    const float* bc1  = (const float*)d_in[20];
    const float* Wc2  = (const float*)d_in[21];
    const float* bc2  = (const float*)d_in[22];
    float* out = (float*)d_out;

    // -------- workspace carve (floats) --------
    float* ws = (float*)d_ws;
    size_t off = 0;
    float* xw1  = ws + off; off += (size_t)NN * F1;   // 12.8M
    float* h1   = ws + off; off += (size_t)NN * F1;   // agg1 -> h1 in place
    float* alS1 = ws + off; off += (size_t)NN * H1;
    float* alD1 = ws + off; off += (size_t)NN * H1;
    int*   m1   = (int*)(ws + off); off += (size_t)NN * H1;
    float* den1 = ws + off; off += (size_t)NN * H1;
    float* exw  = ws + off; off += (size_t)EP * H1;   // shared L1/L2
    float* xw2  = ws + off; off += (size_t)NN * F2;
    float* h2   = ws + off; off += (size_t)NN * F2;   // agg2 -> h2 in place
    float* alS2 = ws + off; off += (size_t)NN;
    float* alD2 = ws + off; off += (size_t)NN;
    int*   m2   = (int*)(ws + off); off += (size_t)NN;
    float* den2 = ws + off; off += (size_t)NN;
    float* sums = ws + off; off += (size_t)GG * F2;
    float* cnt  = ws + off; off += (size_t)GG;

    const int TB = 256;
    const unsigned INT_MIN_BITS = 0x80000000u;

    // ======== Layer 1 ========
    fill_u32_kernel<<<4096, TB, 0, stream>>>((unsigned*)h1,  0u, (long long)NN * F1);
    fill_u32_kernel<<<256,  TB, 0, stream>>>((unsigned*)den1, 0u, (long long)NN * H1);
    fill_u32_kernel<<<256,  TB, 0, stream>>>((unsigned*)m1, INT_MIN_BITS, (long long)NN * H1);

    {   // xw1 = x @ W1   (50000x128 * 128x256)
        int tiles = (NN / 16) * (F1 / 16);
        gemm16_wmma_f32<<<CDIV(tiles, 8), TB, 0, stream>>>(x, W1, xw1, NN, DIN, F1);
    }
    alphas_kernel<H1><<<CDIV(NN * H1, TB), TB, 0, stream>>>(xw1, as1w, ad1w, alS1, alD1);
    edge_max_kernel<H1><<<CDIV(EP * H1, TB), TB, 0, stream>>>(ei, alS1, alD1, m1);
    edge_exp_kernel<H1><<<CDIV(EP * H1, TB), TB, 0, stream>>>(ei, alS1, alD1, m1, den1, exw);
    edge_agg_kernel<H1, HID><<<CDIV((long long)EP * 32, TB), TB, 0, stream>>>(ei, xw1, exw, den1, h1);
    bias_bn_relu_kernel<<<4096, TB, 0, stream>>>(h1, b1, bn1g, bn1b, bn1m, bn1v, F1, (long long)NN * F1);

    // ======== Layer 2 ========
    fill_u32_kernel<<<1024, TB, 0, stream>>>((unsigned*)h2,  0u, (long long)NN * F2);
    fill_u32_kernel<<<128,  TB, 0, stream>>>((unsigned*)den2, 0u, (long long)NN);
    fill_u32_kernel<<<128,  TB, 0, stream>>>((unsigned*)m2, INT_MIN_BITS, (long long)NN);

    {   // xw2 = h1 @ W2   (50000x256 * 256x64)
        int tiles = (NN / 16) * (F2 / 16);
        gemm16_wmma_f32<<<CDIV(tiles, 8), TB, 0, stream>>>(h1, W2, xw2, NN, F1, F2);
    }
    alphas_kernel<1><<<CDIV(NN, TB), TB, 0, stream>>>(xw2, as2w, ad2w, alS2, alD2);
    edge_max_kernel<1><<<CDIV(EP, TB), TB, 0, stream>>>(ei, alS2, alD2, m2);
    edge_exp_kernel<1><<<CDIV(EP, TB), TB, 0, stream>>>(ei, alS2, alD2, m2, den2, exw);
    edge_agg_kernel<1, HID><<<CDIV((long long)EP * 32, TB), TB, 0, stream>>>(ei, xw2, exw, den2, h2);
    bias_bn_relu_kernel<<<2048, TB, 0, stream>>>(h2, b2, bn2g, bn2b, bn2m, bn2v, F2, (long long)NN * F2);

    // ======== Pool + classifier ========
    fill_u32_kernel<<<8, TB, 0, stream>>>((unsigned*)sums, 0u, (long long)(GG * F2 + GG)); // sums + cnt contiguous
    pool_kernel<<<CDIV(NN * F2, TB), TB, 0, stream>>>(h2, batch, sums, cnt);
    classifier_kernel<<<1, 64, 0, stream>>>(sums, cnt, Wc1, bc1, Wc2, bc2, out);
}